// Layer1_73461120631325
// MI455X (gfx1250) — compile-verified
//
#include <hip/hip_runtime.h>

typedef __attribute__((ext_vector_type(2))) float v2f;
typedef __attribute__((ext_vector_type(8))) float v8f;

#define DD     9
#define NP     6
#define KP     12          // K padded to 3 tiles of 4
#define GAMMA  1.3f
#define BETA   2.0f
#define SCALEC 255.0f
#define EPSV   1e-6f

__global__ __launch_bounds__(32) void layer1_kernel(const float* __restrict__ x,
                                                    const float* __restrict__ Hm,
                                                    const float* __restrict__ bm,
                                                    float* __restrict__ out) {
    // Zero-padded WMMA staging tiles (8-byte aligned rows: stride 12 floats)
    __shared__ __align__(16) float s_Ha[16 * KP];   // A: H padded to 16 x 12, row-major [row][k]
    __shared__ __align__(16) float s_Bt[16 * KP];   // B: e padded+transposed, [col n][k]
    __shared__ __align__(16) float s_he[16 * 16];   // D: padded 16x16 result
    __shared__ float s_e[DD * NP];                  // e, compact row-major
    __shared__ float s_cs[NP];                      // column sums of e

    const int tid    = threadIdx.x;   // 0..31, one wave
    const int half   = tid >> 4;      // 0: lanes 0-15, 1: lanes 16-31
    const int lane16 = tid & 15;

    // Zero the padded operand tiles (192 floats each; 6 uniform iterations, no divergence)
    #pragma unroll
    for (int i = 0; i < 6; ++i) {
        s_Ha[tid + 32 * i] = 0.0f;
        s_Bt[tid + 32 * i] = 0.0f;
    }
    __syncthreads();

    // Stage H into padded A tile (81 elements)
    for (int idx = tid; idx < DD * DD; idx += 32) {
        const int r = idx / DD, c = idx % DD;
        s_Ha[r * KP + c] = Hm[idx];
    }

    // Stage 1: yy = (x==0 ? eps : x); e = yy^1.3.
    // Write yy to out[0:54], e to compact LDS and transposed into padded B tile.
    for (int idx = tid; idx < DD * NP; idx += 32) {
        const int r = idx / NP, c = idx % NP;
        float xv = x[idx];
        xv = (xv == 0.0f) ? EPSV : xv;
        out[idx] = xv;                       // yy (L == 1.0)
        const float ev = powf(xv, GAMMA);
        s_e[idx] = ev;
        s_Bt[c * KP + r] = ev;               // B[n][k]
    }
    __syncthreads();

    // Column sums of e (lanes 0..5)
    if (tid < NP) {
        float s = 0.0f;
        #pragma unroll
        for (int k = 0; k < DD; ++k) s += s_e[k * NP + tid];
        s_cs[tid] = s;
    }

    // Stage 2: H @ e via three accumulating V_WMMA_F32_16X16X4_F32 (K = 0..11).
    // Per-lane fragment index: k0 = 4t + 2*half (even), unconditional 8B-aligned LDS loads.
    v8f acc = {};
    #pragma unroll
    for (int t = 0; t < 3; ++t) {
        const int k0 = 4 * t + 2 * half;
        const v2f a  = *(const v2f*)&s_Ha[lane16 * KP + k0];   // A[row=lane16][k0,k0+1]
        const v2f bf = *(const v2f*)&s_Bt[lane16 * KP + k0];   // B[k0,k0+1][col=lane16]
        acc = __builtin_amdgcn_wmma_f32_16x16x4_f32(
            /*neg_a=*/false, a, /*neg_b=*/false, bf,
            /*c_mod=*/(short)0, acc, /*reuse_a=*/false, /*reuse_b=*/false);
    }

    // Harvest D unconditionally into padded 16x16 tile:
    // VGPR v -> row (v + 8*half), col = lane16. All 256 slots covered.
    #pragma unroll
    for (int v = 0; v < 8; ++v) {
        s_he[(v + 8 * half) * 16 + lane16] = acc[v];
    }
    __syncthreads();

    // Stage 3: denom = b + H@e ; K = 255*(b + (2/9)*colsum + 1) ; xim = K*e/denom
    for (int idx = tid; idx < DD * NP; idx += 32) {
        const int r = idx / NP, c = idx % NP;
        const float e  = s_e[idx];
        const float bv = bm[idx];
        const float denom = bv + s_he[r * 16 + c];
        const float Kv = SCALEC * (bv + (BETA / (float)DD) * s_cs[c] + 1.0f);
        out[DD * NP + idx] = Kv * e / denom;
    }
}

extern "C" void kernel_launch(void* const* d_in, const int* in_sizes, int n_in,
                              void* d_out, int out_size, void* d_ws, size_t ws_size,
                              hipStream_t stream) {
    const float* x  = (const float*)d_in[0];
    const float* Hm = (const float*)d_in[1];
    const float* bm = (const float*)d_in[2];
    float* out = (float*)d_out;
    layer1_kernel<<<1, 32, 0, stream>>>(x, Hm, bm, out);
}